// InverseRealSHT_39298950759023
// MI455X (gfx1250) — compile-verified
//
#include <hip/hip_runtime.h>

// ---------------------------------------------------------------------------
// Inverse real SHT on gfx1250:
//   stage T: transpose x[bc,l,m] -> XT[2m+ri][bc][l]            (coalesced)
//   stage B: build iDFT basis  Bmat[2m+ri][n]                   (tiny)
//   stage 1: per-m GEMM  Z[2m+ri][bc*360+k] = XT[2m+ri] @ pct[m]   (WMMA f32)
//   stage 2: GEMM        out[bck][n] = Z^T @ Bmat                  (WMMA f32)
// All matrix math uses V_WMMA_F32_16X16X4_F32 (fp32 in, fp32 acc).
// ---------------------------------------------------------------------------

typedef __attribute__((ext_vector_type(2))) float v2f;
typedef __attribute__((ext_vector_type(8))) float v8f;

#define NLAT   360
#define NLON   720
#define LMAX   360
#define MMAX   361
#define BC     256                 // B*C = 4*64
#define MROWS  (BC * NLAT)         // 92160 output rows (bck)
#define K2     (2 * MMAX)          // 722 (re/im interleaved spectral rows)

#define APITCH 36                  // LDS pitch for 64x32 A tiles (bank-safe)
#define BPITCH 136                 // LDS pitch for 32x128 B tiles (bank-safe)

__device__ __forceinline__ v8f v8f_zero() {
    v8f z;
#pragma unroll
    for (int i = 0; i < 8; ++i) z[i] = 0.0f;
    return z;
}

__device__ __forceinline__ v8f wmma_f32(v2f a, v2f b, v8f c) {
    // D = A(16x4,f32) x B(4x16,f32) + C(16x16,f32)
    return __builtin_amdgcn_wmma_f32_16x16x4_f32(
        /*neg_a=*/false, a, /*neg_b=*/false, b,
        /*c_mod=*/(short)0, c, /*reuse_a=*/false, /*reuse_b=*/false);
}

// ---------------------------------------------------------------------------
// Stage T: x_{re,im}[bc][l][m]  ->  XT[(2m+ri)][bc][l]   (tiled LDS transpose)
// grid: (12 l-tiles, 12 m-tiles, 512 = bc*2), block (32,8)
// ---------------------------------------------------------------------------
__global__ void sht_transpose(const float* __restrict__ xre,
                              const float* __restrict__ xim,
                              float* __restrict__ XT) {
    __shared__ float tile[32][33];
    const int bc = blockIdx.z >> 1;
    const int ri = blockIdx.z & 1;
    const float* __restrict__ src = ri ? xim : xre;
    const int l0 = blockIdx.x * 32;
    const int m0 = blockIdx.y * 32;
    const int tx = threadIdx.x;           // 0..31
    const int ty = threadIdx.y;           // 0..7
    const size_t base = (size_t)bc * LMAX * MMAX;

#pragma unroll
    for (int j = 0; j < 4; ++j) {         // read contiguous in m
        const int l = l0 + ty + j * 8;
        const int m = m0 + tx;
        float v = 0.0f;
        if (l < LMAX && m < MMAX) v = src[base + (size_t)l * MMAX + m];
        tile[ty + j * 8][tx] = v;         // tile[l_loc][m_loc]
    }
    __syncthreads();
#pragma unroll
    for (int j = 0; j < 4; ++j) {         // write contiguous in l
        const int m = m0 + ty + j * 8;
        const int l = l0 + tx;
        if (m < MMAX && l < LMAX)
            XT[((size_t)(2 * m + ri) * BC + bc) * LMAX + l] = tile[tx][ty + j * 8];
    }
}

// ---------------------------------------------------------------------------
// Stage B: iDFT basis. Bmat[2m+0][n] = w_m cos(2pi m n/720)
//                      Bmat[2m+1][n] = -w_m sin(2pi m n/720)
// w_m = 1 for m in {0, 360}, else 2 (Hermitian fold; norm='forward' -> no 1/N)
// ---------------------------------------------------------------------------
__global__ void sht_basis(float* __restrict__ Bmat) {
    const int idx = blockIdx.x * blockDim.x + threadIdx.x;
    if (idx >= K2 * NLON) return;
    const int k2 = idx / NLON;
    const int n  = idx - k2 * NLON;
    const int m  = k2 >> 1;
    const int ri = k2 & 1;
    const float w  = (m == 0 || m == NLON / 2) ? 1.0f : 2.0f;
    const int   ph = (m * n) % NLON;                     // exact integer phase
    const float th = (float)ph * (6.283185307179586f / (float)NLON);
    float s, c;
    __sincosf(th, &s, &c);
    Bmat[idx] = ri ? (-w * s) : (w * c);
}

// ---------------------------------------------------------------------------
// Stage 1: batched-over-m GEMM (re & im share the pct B-tile).
//   O_m[bc][k] = sum_l XT[2m+ri][bc][l] * pct[m][l][k],  l >= (m & ~31) only.
// grid: (3 k-tiles x128, 4 bc-tiles x64, 361 m), block 256 (8 waves).
// Wave tile 32x32 (2x2 WMMA frags) for each of re/im -> 64 acc VGPRs.
// ---------------------------------------------------------------------------
__global__ __launch_bounds__(256)
void sht_legendre_wmma(const float* __restrict__ XT,
                       const float* __restrict__ pct,
                       float* __restrict__ Z) {
    __shared__ float Bs[32 * BPITCH];
    __shared__ float Are[64 * APITCH];
    __shared__ float Aim[64 * APITCH];

    const int m    = blockIdx.z;
    const int k0   = blockIdx.x * 128;
    const int bc0  = blockIdx.y * 64;
    const int tid  = threadIdx.x;
    const int lane = tid & 31;
    const int wave = tid >> 5;
    const int wM   = (wave & 1) * 32;
    const int wN   = (wave >> 1) * 32;
    const int half = lane >> 4;           // K-pair select
    const int lr   = lane & 15;

    v8f acc_re[2][2], acc_im[2][2];
#pragma unroll
    for (int i = 0; i < 2; ++i)
#pragma unroll
        for (int j = 0; j < 2; ++j) { acc_re[i][j] = v8f_zero(); acc_im[i][j] = v8f_zero(); }

    const size_t pbase = (size_t)m * LMAX * NLAT;
    const size_t abre  = (size_t)(2 * m) * BC * LMAX;
    const size_t abim  = abre + (size_t)BC * LMAX;

    for (int l0 = (m & ~31); l0 < LMAX; l0 += 32) {   // triangular skip: l >= m
        __syncthreads();
        for (int i = tid; i < 32 * 128; i += 256) {   // pct tile (shared by re/im)
            const int r = i >> 7, c = i & 127;
            const int l = l0 + r, k = k0 + c;
            float v = 0.0f;
            if (l < LMAX && k < NLAT) v = pct[pbase + (size_t)l * NLAT + k];
            Bs[r * BPITCH + c] = v;
        }
        for (int i = tid; i < 64 * 32; i += 256) {    // A tiles (re & im)
            const int r = i >> 5, c = i & 31;
            const int l = l0 + c;
            float vre = 0.0f, vim = 0.0f;
            if (l < LMAX) {
                const size_t off = (size_t)(bc0 + r) * LMAX + l;
                vre = XT[abre + off];
                vim = XT[abim + off];
            }
            Are[r * APITCH + c] = vre;
            Aim[r * APITCH + c] = vim;
        }
        __syncthreads();

#pragma unroll
        for (int kk = 0; kk < 32; kk += 4) {
            const int ka = kk + 2 * half;
            v2f bf[2];
#pragma unroll
            for (int nb = 0; nb < 2; ++nb) {
                const int nc = wN + nb * 16 + lr;
                bf[nb].x = Bs[ka * BPITCH + nc];
                bf[nb].y = Bs[(ka + 1) * BPITCH + nc];
            }
#pragma unroll
            for (int ma = 0; ma < 2; ++ma) {
                const int ao = (wM + ma * 16 + lr) * APITCH + ka;
                v2f are, aim;
                are.x = Are[ao]; are.y = Are[ao + 1];
                aim.x = Aim[ao]; aim.y = Aim[ao + 1];
#pragma unroll
                for (int nb = 0; nb < 2; ++nb) {
                    acc_re[ma][nb] = wmma_f32(are, bf[nb], acc_re[ma][nb]);
                    acc_im[ma][nb] = wmma_f32(aim, bf[nb], acc_im[ma][nb]);
                }
            }
        }
    }

    const size_t zre = (size_t)(2 * m) * MROWS;
    const size_t zim = zre + MROWS;
#pragma unroll
    for (int ma = 0; ma < 2; ++ma)
#pragma unroll
        for (int nb = 0; nb < 2; ++nb) {
            const int kcol = k0 + wN + nb * 16 + lr;
            if (kcol < NLAT) {
#pragma unroll
                for (int v = 0; v < 8; ++v) {
                    const int bc = bc0 + wM + ma * 16 + half * 8 + v;
                    const size_t col = (size_t)bc * NLAT + kcol;
                    Z[zre + col] = acc_re[ma][nb][v];
                    Z[zim + col] = acc_im[ma][nb][v];
                }
            }
        }
}

// ---------------------------------------------------------------------------
// Stage 2: out[92160 x 720] = Z^T (K-major A) @ Bmat, K = 722 (zero-clamped).
// grid: (6 n-tiles x128, 1440 row-tiles x64), block 256 (8 waves).
// ---------------------------------------------------------------------------
__global__ __launch_bounds__(256)
void sht_idft_wmma(const float* __restrict__ Z,
                   const float* __restrict__ Bmat,
                   float* __restrict__ out) {
    __shared__ float Bs[32 * BPITCH];
    __shared__ float As[64 * APITCH];

    const int n0   = blockIdx.x * 128;
    const int row0 = blockIdx.y * 64;
    const int tid  = threadIdx.x;
    const int lane = tid & 31;
    const int wave = tid >> 5;
    const int wM   = (wave & 1) * 32;
    const int wN   = (wave >> 1) * 32;
    const int half = lane >> 4;
    const int lr   = lane & 15;

    v8f acc[2][2];
#pragma unroll
    for (int i = 0; i < 2; ++i)
#pragma unroll
        for (int j = 0; j < 2; ++j) acc[i][j] = v8f_zero();

    for (int kb = 0; kb < K2; kb += 32) {
        __syncthreads();
        for (int i = tid; i < 32 * 128; i += 256) {   // basis tile
            const int r = i >> 7, c = i & 127;
            const int kk2 = kb + r, n = n0 + c;
            float v = 0.0f;
            if (kk2 < K2 && n < NLON) v = Bmat[(size_t)kk2 * NLON + n];
            Bs[r * BPITCH + c] = v;
        }
        for (int i = tid; i < 32 * 64; i += 256) {    // Z tile (K-major -> LDS^T)
            const int r = i >> 6, c = i & 63;
            const int kk2 = kb + r;
            float v = 0.0f;
            if (kk2 < K2) v = Z[(size_t)kk2 * MROWS + (row0 + c)];
            As[c * APITCH + r] = v;
        }
        __syncthreads();

#pragma unroll
        for (int kk = 0; kk < 32; kk += 4) {
            const int ka = kk + 2 * half;
            v2f bf[2];
#pragma unroll
            for (int nb = 0; nb < 2; ++nb) {
                const int nc = wN + nb * 16 + lr;
                bf[nb].x = Bs[ka * BPITCH + nc];
                bf[nb].y = Bs[(ka + 1) * BPITCH + nc];
            }
#pragma unroll
            for (int ma = 0; ma < 2; ++ma) {
                const int ao = (wM + ma * 16 + lr) * APITCH + ka;
                v2f a; a.x = As[ao]; a.y = As[ao + 1];
#pragma unroll
                for (int nb = 0; nb < 2; ++nb)
                    acc[ma][nb] = wmma_f32(a, bf[nb], acc[ma][nb]);
            }
        }
    }

#pragma unroll
    for (int ma = 0; ma < 2; ++ma)
#pragma unroll
        for (int nb = 0; nb < 2; ++nb) {
            const int n = n0 + wN + nb * 16 + lr;
            if (n < NLON) {
#pragma unroll
                for (int v = 0; v < 8; ++v) {
                    const int row = row0 + wM + ma * 16 + half * 8 + v;
                    out[(size_t)row * NLON + n] = acc[ma][nb][v];
                }
            }
        }
}

// ---------------------------------------------------------------------------
extern "C" void kernel_launch(void* const* d_in, const int* in_sizes, int n_in,
                              void* d_out, int out_size, void* d_ws, size_t ws_size,
                              hipStream_t stream) {
    (void)in_sizes; (void)n_in; (void)out_size; (void)ws_size;
    const float* x_re = (const float*)d_in[0];
    const float* x_im = (const float*)d_in[1];
    const float* pct  = (const float*)d_in[2];
    float* out = (float*)d_out;

    float* XT   = (float*)d_ws;                          // [722][256][360]
    float* Zbuf = XT + (size_t)K2 * BC * LMAX;           // [722][92160]
    float* Bm   = Zbuf + (size_t)K2 * MROWS;             // [722][720]

    sht_transpose<<<dim3(12, 12, 2 * BC), dim3(32, 8, 1), 0, stream>>>(x_re, x_im, XT);
    sht_basis<<<(K2 * NLON + 255) / 256, 256, 0, stream>>>(Bm);
    sht_legendre_wmma<<<dim3(3, 4, MMAX), 256, 0, stream>>>(XT, pct, Zbuf);
    sht_idft_wmma<<<dim3(6, MROWS / 64), 256, 0, stream>>>(Zbuf, Bm, out);
}